// MeshLaplacianLoss_old_51857435132213
// MI455X (gfx1250) — compile-verified
//
#include <hip/hip_runtime.h>

#define NUM_POINT 100000
#define NUM_FACES 200000
#define BATCH 4

typedef __attribute__((ext_vector_type(2))) float v2f;
typedef __attribute__((ext_vector_type(8))) float v8f;

// ---------------------------------------------------------------------------
// Kernel 1: zero the workspace accumulators (nn_d + deg) and d_out[0].
// ---------------------------------------------------------------------------
__global__ void lap_zero_kernel(float* __restrict__ ws, long n, float* __restrict__ out) {
  long t = (long)blockIdx.x * blockDim.x + threadIdx.x;
  long s = (long)gridDim.x * blockDim.x;
  for (long i = t; i < n; i += s) ws[i] = 0.0f;
  if (t == 0) out[0] = 0.0f;
}

// ---------------------------------------------------------------------------
// Kernel 2: scatter. One thread per (batch, face). Scatters on d = v1 - v2,
// combining the two neighbor contributions per target vertex into a single
// float atomic per component (9 f32 atomics + 3 deg atomics per face).
// All accumulators live in L2 (192 MB >> 6.4 MB), so these are L2 atomics.
// ---------------------------------------------------------------------------
__global__ void lap_scatter_kernel(const float* __restrict__ v1,
                                   const float* __restrict__ v2,
                                   const int*   __restrict__ faces,
                                   float* __restrict__ nn,
                                   float* __restrict__ deg) {
  int t = blockIdx.x * blockDim.x + threadIdx.x;
  if (t >= BATCH * NUM_FACES) return;
  int b = t / NUM_FACES;
  const int* fp = faces + (long)t * 3;
  long vbase = (long)b * NUM_POINT;
  long i = vbase + fp[0];
  long j = vbase + fp[1];
  long k = vbase + fp[2];

  float di[3], dj[3], dk[3];
#pragma unroll
  for (int c = 0; c < 3; ++c) {
    di[c] = v1[i * 3 + c] - v2[i * 3 + c];
    dj[c] = v1[j * 3 + c] - v2[j * 3 + c];
    dk[c] = v1[k * 3 + c] - v2[k * 3 + c];
  }
#pragma unroll
  for (int c = 0; c < 3; ++c) {
    atomicAdd(&nn[i * 3 + c], dj[c] + dk[c]);
    atomicAdd(&nn[j * 3 + c], di[c] + dk[c]);
    atomicAdd(&nn[k * 3 + c], di[c] + dj[c]);
  }
  atomicAdd(&deg[i], 2.0f);
  atomicAdd(&deg[j], 2.0f);
  atomicAdd(&deg[k], 2.0f);
}

// ---------------------------------------------------------------------------
// Kernel 3: finalize. Per-vertex |nn/deg - d| summed over xyz, then a full-
// precision f32 reduction through V_WMMA_F32_16X16X4_F32 (A = data, B = ones,
// C accumulates). Uniform trip count + value masking keeps EXEC all-ones at
// every WMMA, as the ISA requires.
//
// deg is a small integer-valued float, so a single hardware v_rcp_f32
// (~1 ulp) + 3 multiplies replaces three full IEEE division sequences.
// ---------------------------------------------------------------------------
__device__ __forceinline__ float lap_vertex_term(long n,
                                                 const float* __restrict__ v1,
                                                 const float* __restrict__ v2,
                                                 const float* __restrict__ nn,
                                                 const float* __restrict__ deg) {
  float dg = deg[n];
  dg = dg > 1.0f ? dg : 1.0f;
  float inv = __builtin_amdgcn_rcpf(dg);  // v_rcp_f32, ~1 ulp
  long b3 = n * 3;
  float s = 0.0f;
#pragma unroll
  for (int c = 0; c < 3; ++c) {
    float d   = v1[b3 + c] - v2[b3 + c];
    float lap = fmaf(nn[b3 + c], inv, -d);
    s += fabsf(lap);
  }
  return s;
}

__global__ void lap_finalize_kernel(const float* __restrict__ v1,
                                    const float* __restrict__ v2,
                                    const float* __restrict__ nn,
                                    const float* __restrict__ deg,
                                    float* __restrict__ out) {
  const long total  = (long)BATCH * NUM_POINT;
  const long tid    = (long)blockIdx.x * blockDim.x + threadIdx.x;
  const long stride = (long)gridDim.x * blockDim.x;
  const long step   = 2 * stride;                 // 2 f32 per lane per WMMA (16x4 A tile)
  const long iters  = (total + step - 1) / step;  // uniform across all lanes

  v8f acc = {0.0f, 0.0f, 0.0f, 0.0f, 0.0f, 0.0f, 0.0f, 0.0f};
  v2f bones;
  bones.x = 1.0f;
  bones.y = 1.0f;

  for (long m = 0; m < iters; ++m) {
    long i0 = tid + m * step;
    long i1 = i0 + stride;
    float x = 0.0f, y = 0.0f;
    if (i0 < total) x = lap_vertex_term(i0, v1, v2, nn, deg);
    if (i1 < total) y = lap_vertex_term(i1, v1, v2, nn, deg);
    v2f a;
    a.x = x;
    a.y = y;
    // D[m][n] = sum_k A[m][k] * 1 + C[m][n]  -> every column accumulates the
    // row sums; full f32 precision. Emits v_wmma_f32_16x16x4_f32.
    acc = __builtin_amdgcn_wmma_f32_16x16x4_f32(
        false, a, false, bones, (short)0, acc, false, false);
  }

  // Per-lane: lanes 0-15 hold rows M=0..7 in VGPRs 0..7, lanes 16-31 hold
  // rows M=8..15; every column of D is identical. So grand total per wave is
  // s(lane<16) + s(lane>=16).
  float s = acc[0] + acc[1] + acc[2] + acc[3] + acc[4] + acc[5] + acc[6] + acc[7];
  s += __shfl_xor(s, 16, 32);

  if ((threadIdx.x & 31) == 0) {
    const float inv_elems = 1.0f / ((float)BATCH * (float)NUM_POINT * 3.0f);
    atomicAdd(out, s * inv_elems);
  }
}

// ---------------------------------------------------------------------------
// Launch: zero -> scatter -> finalize, all on `stream` (graph-capture safe).
// Workspace layout: [ nn_d : B*N*3 f32 ][ deg : B*N f32 ]  (~6.4 MB).
// ---------------------------------------------------------------------------
extern "C" void kernel_launch(void* const* d_in, const int* in_sizes, int n_in,
                              void* d_out, int out_size, void* d_ws, size_t ws_size,
                              hipStream_t stream) {
  (void)in_sizes; (void)n_in; (void)out_size; (void)ws_size;

  const float* v1    = (const float*)d_in[0];
  const float* v2    = (const float*)d_in[1];
  const int*   faces = (const int*)d_in[2];
  float*       out   = (float*)d_out;

  float* nn  = (float*)d_ws;
  float* deg = nn + (long)BATCH * NUM_POINT * 3;
  const long zn = (long)BATCH * NUM_POINT * 3 + (long)BATCH * NUM_POINT;

  lap_zero_kernel<<<1024, 256, 0, stream>>>(nn, zn, out);

  const int nfaces = BATCH * NUM_FACES;
  lap_scatter_kernel<<<(nfaces + 255) / 256, 256, 0, stream>>>(v1, v2, faces, nn, deg);

  lap_finalize_kernel<<<512, 256, 0, stream>>>(v1, v2, nn, deg, out);
}